// TorchOperator_66305705115790
// MI455X (gfx1250) — compile-verified
//
#include <hip/hip_runtime.h>
#include <stdint.h>

// Problem geometry (fixed by the reference): arr (4,4096,4096) f32 -> full (4,4098,4098) f32
#define F_   4
#define NX_  4096
#define NY_  4096
#define FW_  (NX_ + 2)          // 4098
#define RTILE 2                 // rows per TDM tile -> 2*4096*4 = 32 KB LDS
#define BULK_BLOCKS (F_ * (NX_ / RTILE))   // 8192 one-wave blocks

typedef unsigned int u32x4 __attribute__((ext_vector_type(4)));
typedef int          i32x4 __attribute__((ext_vector_type(4)));
typedef int          i32x8 __attribute__((ext_vector_type(8)));
typedef float        f4a   __attribute__((ext_vector_type(4), aligned(16)));
typedef float        f4u   __attribute__((ext_vector_type(4), aligned(4)));  // dst rows are only 4B aligned

// Use the Tensor Data Mover when the builtins exist (probe-confirmed on both toolchains).
#if defined(__HIP_DEVICE_COMPILE__) && \
    __has_builtin(__builtin_amdgcn_tensor_load_to_lds) && \
    __has_builtin(__builtin_amdgcn_tensor_store_from_lds) && \
    __has_builtin(__builtin_amdgcn_s_wait_tensorcnt)
#define USE_TDM 1
#else
#define USE_TDM 0
#endif

// Build a 2D TDM descriptor (D# groups 0 and 1) per CDNA5 ISA §8.3/§8.4.
//   group0: [1:0]=count=1, [63:32]=lds byte addr, [120:64]=57b global byte addr, [127:126]=type=2
//   group1: [17:16]=data_size(2 -> 4B), tensor_dim0/1, tile_dim0/1 (tile_dim2=0 => 2D),
//           tensor_dim0_stride (48b, element units). workgroup_mask=0 (not in a cluster).
__device__ __forceinline__ void tdm_make_desc(unsigned long long gaddr,
                                              unsigned lds_addr,
                                              unsigned tile_w, unsigned tile_h,
                                              unsigned long long row_stride_elems,
                                              u32x4* g0, i32x8* g1) {
  (*g0)[0] = 1u;                                                    // count=1, user mode
  (*g0)[1] = lds_addr;                                              // LDS byte address
  (*g0)[2] = (unsigned)(gaddr & 0xFFFFFFFFull);                     // global_addr[31:0]
  (*g0)[3] = (unsigned)((gaddr >> 32) & 0x01FFFFFFull) | (2u << 30); // addr[56:32] | type=2

  const unsigned long long td0 = tile_w;   // tensor_dim0 == tile width (tile never overruns)
  const unsigned long long td1 = tile_h;   // tensor_dim1 == tile height
  (*g1)[0] = (int)(2u << 16);                                            // data_size = 4 bytes
  (*g1)[1] = (int)((td0 & 0xFFFFull) << 16);                             // tensor_dim0[15:0]
  (*g1)[2] = (int)(((td0 >> 16) & 0xFFFFull) | ((td1 & 0xFFFFull) << 16)); // td0[31:16] | td1[15:0]
  (*g1)[3] = (int)(((td1 >> 16) & 0xFFFFull) |
                   ((unsigned long long)(tile_w & 0xFFFFu) << 16));      // td1[31:16] | tile_dim0
  (*g1)[4] = (int)(tile_h & 0xFFFFu);                                    // tile_dim1 | tile_dim2=0
  (*g1)[5] = (int)(row_stride_elems & 0xFFFFFFFFull);                    // dim0_stride[31:0]
  (*g1)[6] = (int)((row_stride_elems >> 32) & 0xFFFFull);                // dim0_stride[47:32]
  (*g1)[7] = 0;                                                          // dim1_stride (unused, 2D)
}

// Bulk interior copy: full[f, 1+i, 1:NY+1] = arr[f, i, :].
// One wave per block; tensor ops ignore EXEC, so a single wave issues each DMA exactly once.
__global__ __launch_bounds__(32) void bulk_kernel(const float* __restrict__ in,
                                                  float* __restrict__ out) {
  const int per_f = NX_ / RTILE;
  const int f  = (int)blockIdx.x / per_f;
  const int r0 = ((int)blockIdx.x % per_f) * RTILE;
  const size_t in_off  = (size_t)f * NX_ * NY_ + (size_t)r0 * NY_;
  const size_t out_off = (size_t)f * FW_ * FW_ + (size_t)(r0 + 1) * FW_ + 1;
#if USE_TDM
  __shared__ float tile[RTILE * NY_];   // 32 KB staging tile
  u32x4 g0l, g0s; i32x8 g1l, g1s;
  const unsigned lds_addr = (unsigned)(uintptr_t)(&tile[0]);  // low 32b of generic = LDS byte addr
  tdm_make_desc((unsigned long long)(uintptr_t)(in + in_off), lds_addr,
                NY_, RTILE, (unsigned long long)NY_, &g0l, &g1l);     // src: stride 4096
  tdm_make_desc((unsigned long long)(uintptr_t)(out + out_off), lds_addr,
                NY_, RTILE, (unsigned long long)FW_, &g0s, &g1s);     // dst: stride 4098, +1 offset
  i32x4 z4 = {0, 0, 0, 0};
#if __clang_major__ >= 23
  i32x8 z8 = {0, 0, 0, 0, 0, 0, 0, 0};
  __builtin_amdgcn_tensor_load_to_lds(g0l, g1l, z4, z4, z8, 0);
  __builtin_amdgcn_s_wait_tensorcnt(0);
  __builtin_amdgcn_tensor_store_from_lds(g0s, g1s, z4, z4, z8, 0);
#else
  __builtin_amdgcn_tensor_load_to_lds(g0l, g1l, z4, z4, 0);
  __builtin_amdgcn_s_wait_tensorcnt(0);
  __builtin_amdgcn_tensor_store_from_lds(g0s, g1s, z4, z4, 0);
#endif
  __builtin_amdgcn_s_wait_tensorcnt(0);
#else
  // Fallback: direct vectorized copy (b128 loads 16B-aligned; b128 stores 4B-aligned, legal
  // for global memory in DWORD alignment mode).
  const int nchunk = RTILE * (NY_ / 4);
  for (int k = (int)threadIdx.x; k < nchunk; k += 32) {
    const int row = k / (NY_ / 4);
    const int c4  = (k % (NY_ / 4)) * 4;
    const f4a v = *(const f4a*)(in + in_off + (size_t)row * NY_ + c4);
    *(f4u*)(out + out_off + (size_t)row * FW_ + c4) = (f4u)v;
  }
#endif
}

// Ghost lines + corners. bc arrays are (2,2) row-major: [axis0 lo, axis0 hi, axis1 lo, axis1 hi].
__global__ __launch_bounds__(256) void edge_kernel(const float* __restrict__ in,
                                                   const float* __restrict__ bc_c,
                                                   const float* __restrict__ bc_f,
                                                   float* __restrict__ out) {
  const int j = (int)(blockIdx.x * blockDim.x + threadIdx.x);
  const int f = (int)blockIdx.y;
  if (j >= NX_) return;
  const size_t ib = (size_t)f * NX_ * NY_;
  const size_t ob = (size_t)f * FW_ * FW_;
  const float c0l = bc_c[0], c0h = bc_c[1], c1l = bc_c[2], c1h = bc_c[3];
  const float f0l = bc_f[0], f0h = bc_f[1], f1l = bc_f[2], f1h = bc_f[3];
  // axis 0 (row) ghosts: rows 0 and FW-1, interior columns 1..NX
  out[ob + (size_t)(j + 1)]                   = c0l + f0l * in[ib + (size_t)j];
  out[ob + (size_t)(FW_ - 1) * FW_ + (j + 1)] = c0h + f0h * in[ib + (size_t)(NX_ - 1) * NY_ + j];
  // axis 1 (column) ghosts: cols 0 and FW-1, interior rows 1..NX
  out[ob + (size_t)(j + 1) * FW_]             = c1l + f1l * in[ib + (size_t)j * NY_];
  out[ob + (size_t)(j + 1) * FW_ + (FW_ - 1)] = c1h + f1h * in[ib + (size_t)j * NY_ + (NY_ - 1)];
  if (j == 0) {  // corners stay zero (output buffer is poisoned, so write them explicitly)
    out[ob + 0] = 0.0f;
    out[ob + (size_t)(FW_ - 1)] = 0.0f;
    out[ob + (size_t)(FW_ - 1) * FW_] = 0.0f;
    out[ob + (size_t)(FW_ - 1) * FW_ + (FW_ - 1)] = 0.0f;
  }
}

extern "C" void kernel_launch(void* const* d_in, const int* in_sizes, int n_in,
                              void* d_out, int out_size, void* d_ws, size_t ws_size,
                              hipStream_t stream) {
  (void)in_sizes; (void)n_in; (void)out_size; (void)d_ws; (void)ws_size;
  const float* arr  = (const float*)d_in[0];
  const float* bc_c = (const float*)d_in[1];  // 4 floats
  const float* bc_f = (const float*)d_in[2];  // 4 floats
  float* out = (float*)d_out;                 // 4 * 4098 * 4098 floats

  bulk_kernel<<<dim3(BULK_BLOCKS), dim3(32), 0, stream>>>(arr, out);
  edge_kernel<<<dim3(NX_ / 256, F_), dim3(256), 0, stream>>>(arr, bc_c, bc_f, out);
}